// MemoryEfficientDiceLoss_9182640079166
// MI455X (gfx1250) — compile-verified
//
#include <hip/hip_runtime.h>
#include <hip/hip_bf16.h>

// MemoryEfficientDiceLoss for (B=2, C=16, D=64, H=128, W=128), targets int.
// Memory-bound streaming softmax + 3 segmented reductions -> scalar loss.

typedef float v2f __attribute__((ext_vector_type(2)));
typedef float v8f __attribute__((ext_vector_type(8)));
typedef float fl4 __attribute__((ext_vector_type(4)));
typedef int   i4  __attribute__((ext_vector_type(4)));

#define CCLS 16
#define THREADS 256
#define VOX_PER_THREAD 8
#define VOX_PER_BLOCK (THREADS * VOX_PER_THREAD)  // 2048

__global__ __launch_bounds__(THREADS) void dice_partial_kernel(
    const float* __restrict__ logits,   // [B][C][N]
    const int*   __restrict__ targets,  // [B][N]
    float*       __restrict__ ws,       // [B][C][3] = {probs_sum, intersection, count}
    int N)                              // voxels per batch image (D*H*W)
{
  const int tid = threadIdx.x;
  const long long base = (long long)blockIdx.x * VOX_PER_BLOCK;
  const int b  = (int)(base / N);               // block never straddles a batch (N % 2048 == 0)
  const int n0 = (int)(base - (long long)b * N);
  const float* lg = logits + (size_t)b * CCLS * (size_t)N;
  const int*   tg = targets + (size_t)b * (size_t)N;

  float psum[CCLS], inter[CCLS], cnt[CCLS];
  #pragma unroll
  for (int c = 0; c < CCLS; ++c) { psum[c] = 0.f; inter[c] = 0.f; cnt[c] = 0.f; }

  #pragma unroll
  for (int it = 0; it < VOX_PER_THREAD / 4; ++it) {
    const int n = n0 + it * (THREADS * 4) + tid * 4;

    // Coalesced b128 NT loads: one float4 per class plane (class stride = N).
    float x[CCLS][4];
    #pragma unroll
    for (int c = 0; c < CCLS; ++c) {
      fl4 v = __builtin_nontemporal_load((const fl4*)(lg + (size_t)c * N + n));
      x[c][0] = v.x; x[c][1] = v.y; x[c][2] = v.z; x[c][3] = v.w;
    }
    i4 tv = __builtin_nontemporal_load((const i4*)(tg + n));
    int t4[4] = { tv.x, tv.y, tv.z, tv.w };

    #pragma unroll
    for (int j = 0; j < 4; ++j) {
      float m = x[0][j];
      #pragma unroll
      for (int c = 1; c < CCLS; ++c) m = fmaxf(m, x[c][j]);
      float p[CCLS];
      float s = 0.f;
      #pragma unroll
      for (int c = 0; c < CCLS; ++c) { p[c] = __expf(x[c][j] - m); s += p[c]; }
      const float inv = __builtin_amdgcn_rcpf(s);
      const int t = t4[j];
      #pragma unroll
      for (int c = 0; c < CCLS; ++c) {
        const float pc = p[c] * inv;
        psum[c] += pc;
        const bool e = (t == c);
        inter[c] += e ? pc  : 0.f;
        cnt[c]   += e ? 1.f : 0.f;
      }
    }
  }

  // ---- Block reduction: LDS float atomics into per-wave-group slots ----
  __shared__ float acc2[3][CCLS][8];   // [quantity][class][wave-group]
  for (int i = tid; i < 3 * CCLS * 8; i += THREADS) ((float*)acc2)[i] = 0.f;
  __syncthreads();
  const int g = tid >> 5;              // wave id within block (wave32)
  #pragma unroll
  for (int c = 0; c < CCLS; ++c) {
    atomicAdd(&acc2[0][c][g], psum[c]);
    atomicAdd(&acc2[1][c][g], inter[c]);
    atomicAdd(&acc2[2][c][g], cnt[c]);
  }
  __syncthreads();

  // ---- Wave 0: per-class sum of the 8 wave partials via one WMMA per quantity.
  // A = ones(16x4)  =>  D[m][n] = sum over all K-slots of B column n.
  // B column index == lane & 15, so lane (h,c) loads class-c partials; exact
  // K-slot ordering is irrelevant because every slot is summed.
  if (tid < 32) {
    const int c = tid & 15;
    const int h = tid >> 4;
#if __has_builtin(__builtin_amdgcn_wmma_f32_16x16x4_f32)
    const v2f a = { 1.0f, 1.0f };
    #pragma unroll
    for (int q = 0; q < 3; ++q) {
      v2f bv;
      bv.x = acc2[q][c][2 * h + 0] + acc2[q][c][2 * h + 4];
      bv.y = acc2[q][c][2 * h + 1] + acc2[q][c][2 * h + 5];
      v8f cz = { 0.f, 0.f, 0.f, 0.f, 0.f, 0.f, 0.f, 0.f };
      v8f d = __builtin_amdgcn_wmma_f32_16x16x4_f32(
          /*neg_a=*/false, a, /*neg_b=*/false, bv,
          /*c_mod=*/(short)0, cz, /*reuse_a=*/false, /*reuse_b=*/false);
      if (tid < 16)  // D VGPR0, lanes 0-15: row M=0, column N=lane -> class totals
        atomicAdd(&ws[((size_t)b * CCLS + c) * 3 + q], d[0]);
    }
#else
    if (tid < 16) {
      #pragma unroll
      for (int q = 0; q < 3; ++q) {
        float s = 0.f;
        #pragma unroll
        for (int gg = 0; gg < 8; ++gg) s += acc2[q][c][gg];
        atomicAdd(&ws[((size_t)b * CCLS + c) * 3 + q], s);
      }
    }
#endif
  }
}

__global__ void dice_zero_ws(float* __restrict__ ws) {
  if (threadIdx.x < 2 * CCLS * 3) ws[threadIdx.x] = 0.f;
}

__global__ void dice_finalize_kernel(const float* __restrict__ ws,
                                     float* __restrict__ out) {
  // 32 threads: lane -> (b = lane>>4, c = lane&15)
  const int l = threadIdx.x;
  const int b = l >> 4;
  const int c = l & 15;
  const float I = ws[((size_t)b * CCLS + c) * 3 + 0 * 0 + 1];      // intersection (q=1)
  const float P = ws[((size_t)b * CCLS + c) * 3 + 0];              // probs_sum   (q=0)
  const float T = ws[((size_t)b * CCLS + c) * 3 + 2];              // count       (q=2)
  const float dice = (2.0f * I + 1.0f) / (P + T + 1.0f);
  float v = (c == 0) ? 0.0f : dice;   // IGNORE_INDEX = 0 excluded from mean
  #pragma unroll
  for (int off = 16; off > 0; off >>= 1) v += __shfl_xor(v, off, 32);
  if (l == 0) out[0] = 1.0f - v * (1.0f / 30.0f);   // B*(C-1) = 30
}

extern "C" void kernel_launch(void* const* d_in, const int* in_sizes, int n_in,
                              void* d_out, int out_size, void* d_ws, size_t ws_size,
                              hipStream_t stream) {
  const float* logits  = (const float*)d_in[0];
  const int*   targets = (const int*)d_in[1];
  float*       out     = (float*)d_out;
  float*       ws      = (float*)d_ws;

  const int B = 2;
  const int N = in_sizes[1] / B;                 // 64*128*128 = 1,048,576
  const int blocks = (B * N) / VOX_PER_BLOCK;    // 1024

  dice_zero_ws<<<1, 128, 0, stream>>>(ws);
  dice_partial_kernel<<<blocks, THREADS, 0, stream>>>(logits, targets, ws, N);
  dice_finalize_kernel<<<1, 32, 0, stream>>>(ws, out);
}